// ModelLossSemsegGatedCRF3D22D_88235808129167
// MI455X (gfx1250) — compile-verified
//
#include <hip/hip_runtime.h>
#include <stdint.h>

// ---------------- CDNA5 TDM descriptor helpers ----------------
typedef unsigned int v4u __attribute__((ext_vector_type(4)));
typedef int          v4i __attribute__((ext_vector_type(4)));
typedef int          v8i __attribute__((ext_vector_type(8)));

#define N_  2
#define C_  2
#define H_  64
#define W_  64
#define D_  64
#define R_  3
#define TH  8          // tile height in pixels
#define TW  4          // tile width in pixels
#define HALO_H (TH + 2 * R_)   // 14
#define HALO_W (TW + 2 * R_)   // 10

__device__ __forceinline__ void put_bits(unsigned* w, int lo, int n, unsigned long long v) {
  int i = lo >> 5, sh = lo & 31;
  unsigned long long cur = v << sh;
  w[i] |= (unsigned)cur;
  if (sh + n > 32) w[i + 1] |= (unsigned)(cur >> 32);
}

// Issue one TENSOR_LOAD_TO_LDS for a tile:
//   X = d (32 contiguous elems), Y = w (wlen, stride dim0_stride),
//   Z = h (hlen, stride dim1_stride), ZZ = c (cdim, stride dim2_stride; 0 -> 3D).
__device__ __forceinline__ void tdm_load_tile(unsigned lds_addr, unsigned long long gaddr,
                                              int wlen, int hlen, int cdim,
                                              unsigned long long dim0_stride,
                                              unsigned long long dim1_stride,
                                              unsigned long long dim2_stride) {
  unsigned g0[4], g1[8] = {0,0,0,0,0,0,0,0}, g2[4] = {0,0,0,0}, g3[4] = {0,0,0,0};
  g0[0] = 1u;                                   // count = 1 valid descriptor
  g0[1] = lds_addr;                             // LDS byte address
  g0[2] = (unsigned)gaddr;                      // global_addr[31:0]
  g0[3] = (unsigned)((gaddr >> 32) & 0x1FFFFFFull) | (2u << 30);  // addr[56:32] | type=2

  put_bits(g1, 16, 2, 2);                       // data_size = 4 bytes
  put_bits(g1, 48, 32, 32);                     // tensor_dim0 = 32 (== tile, no OOB)
  put_bits(g1, 80, 32, (unsigned)wlen);         // tensor_dim1
  put_bits(g1, 112, 16, 32);                    // tile_dim0 = 32 (d chunk)
  put_bits(g1, 128, 16, (unsigned)wlen);        // tile_dim1
  put_bits(g1, 144, 16, (unsigned)hlen);        // tile_dim2
  put_bits(g1, 160, 48, dim0_stride);           // stride between w steps (elems)
  put_bits(g1, 208, 48, dim1_stride);           // stride between h steps (elems)

  put_bits(g2, 0, 32, (unsigned)hlen);          // tensor_dim2
  if (cdim > 1) {
    put_bits(g2, 32, 32, (unsigned)cdim);       // tensor_dim3
    put_bits(g2, 64, 48, dim2_stride);          // stride between c steps (elems)
    put_bits(g2, 112, 16, (unsigned)cdim);      // tile_dim3
  }

  v4u G0 = {g0[0], g0[1], g0[2], g0[3]};
  v8i G1 = {(int)g1[0], (int)g1[1], (int)g1[2], (int)g1[3],
            (int)g1[4], (int)g1[5], (int)g1[6], (int)g1[7]};
  v4i G2 = {(int)g2[0], (int)g2[1], (int)g2[2], (int)g2[3]};
  v4i G3 = {(int)g3[0], (int)g3[1], (int)g3[2], (int)g3[3]};
#if __clang_major__ >= 23
  v8i GZ = {0, 0, 0, 0, 0, 0, 0, 0};
  __builtin_amdgcn_tensor_load_to_lds(G0, G1, G2, G3, GZ, 0);
#else
  __builtin_amdgcn_tensor_load_to_lds(G0, G1, G2, G3, 0);
#endif
}

// ---------------- main CRF kernel ----------------
// grid: (W_/TW=16, H_/TH=8, N_*2=4)  block: 256 threads (8 waves)
// lane(0..31) = d within 32-wide d-chunk -> coalesced & LDS conflict-free
__global__ __launch_bounds__(256) void gatedcrf_kernel(const float* __restrict__ yg,
                                                       const float* __restrict__ sg,
                                                       float* __restrict__ ws) {
  __shared__ float s_s[HALO_H * HALO_W * 32];            // sample tile   (17.5 KB)
  __shared__ float s_y[2 * HALO_H * HALO_W * 32];        // y tile, c=0,1 (35 KB)
  __shared__ float red[256];

  const int tid = threadIdx.x;
  const int n  = (int)blockIdx.z >> 1;
  const int d0 = ((int)blockIdx.z & 1) * 32;
  const int h0 = (int)blockIdx.y * TH;
  const int w0 = (int)blockIdx.x * TW;

  const int hlo = max(h0 - R_, 0), hhi = min(h0 + TH + R_, H_);
  const int wlo = max(w0 - R_, 0), whi = min(w0 + TW + R_, W_);
  const int hlen = hhi - hlo, wlen = whi - wlo;
  const int cstride_lds = hlen * wlen * 32;              // c=1 offset in s_y

  if (tid == 0) {
    // sample: layout [n][1][h][w][d]
    unsigned long long ga_s = (unsigned long long)(uintptr_t)
        (sg + ((size_t)n * H_ * W_ * D_ + (size_t)hlo * W_ * D_ + (size_t)wlo * D_ + d0));
    tdm_load_tile((unsigned)(uintptr_t)&s_s[0], ga_s, wlen, hlen, /*cdim=*/0,
                  /*w*/D_, /*h*/(unsigned long long)W_ * D_, 0);
    // y: layout [n][c][h][w][d], 4D tile over (d,w,h,c)
    unsigned long long ga_y = (unsigned long long)(uintptr_t)
        (yg + ((size_t)n * C_ * H_ * W_ * D_ + (size_t)hlo * W_ * D_ + (size_t)wlo * D_ + d0));
    tdm_load_tile((unsigned)(uintptr_t)&s_y[0], ga_y, wlen, hlen, /*cdim=*/C_,
                  /*w*/D_, /*h*/(unsigned long long)W_ * D_,
                  /*c*/(unsigned long long)H_ * W_ * D_);
    __builtin_amdgcn_s_wait_tensorcnt(0);
  }
  __syncthreads();   // TDM data visible to all 8 waves

  // spatial affinity exp(-(dy^2+dx^2)/72), center unused
  const float WSP[7][7] = {
    {0.7788008f, 0.8348062f, 0.8703247f, 0.8824969f, 0.8703247f, 0.8348062f, 0.7788008f},
    {0.8348062f, 0.8948393f, 0.9329120f, 0.9459595f, 0.9329120f, 0.8948393f, 0.8348062f},
    {0.8703247f, 0.9329120f, 0.9726044f, 0.9862071f, 0.9726044f, 0.9329120f, 0.8703247f},
    {0.8824969f, 0.9459595f, 0.9862071f, 0.0f      , 0.9862071f, 0.9459595f, 0.8824969f},
    {0.8703247f, 0.9329120f, 0.9726044f, 0.9862071f, 0.9726044f, 0.9329120f, 0.8703247f},
    {0.8348062f, 0.8948393f, 0.9329120f, 0.9459595f, 0.9329120f, 0.8948393f, 0.8348062f},
    {0.7788008f, 0.8348062f, 0.8703247f, 0.8824969f, 0.8703247f, 0.8348062f, 0.7788008f}};

  const int dl = tid & 31;          // d lane
  const int pbase = tid >> 5;       // 0..7
  float tsum = 0.0f;

  for (int p = pbase; p < TH * TW; p += 8) {
    const int ph = p >> 2, pw = p & 3;
    const int h = h0 + ph, w = w0 + pw;
    const int lb = ((h - hlo) * wlen + (w - wlo)) * 32 + dl;
    const float sc  = s_s[lb];
    const float y0c = s_y[lb];
    const float y1c = s_y[cstride_lds + lb];
    // zero-padded neighbor: k = exp(-0.5*||f_center||^2)
    const float cn  = (float)(h * h + w * w) * (1.0f / 36.0f) + 100.0f * sc * sc;
    const float kout = __expf(-0.5f * cn);

    float acck = 0.0f, a0 = 0.0f, a1 = 0.0f;
#pragma unroll
    for (int dy = -R_; dy <= R_; ++dy) {
#pragma unroll
      for (int dx = -R_; dx <= R_; ++dx) {
        if (dy == 0 && dx == 0) continue;          // self-affinity zeroed
        const int hh = h + dy, ww = w + dx;
        if ((unsigned)hh < (unsigned)H_ && (unsigned)ww < (unsigned)W_) {
          const int li = ((hh - hlo) * wlen + (ww - wlo)) * 32 + dl;
          const float dsv = s_s[li] - sc;
          const float k = WSP[dy + R_][dx + R_] * __expf(-50.0f * dsv * dsv);
          acck += k;
          a0 += k * s_y[li];
          a1 += k * s_y[cstride_lds + li];
        } else {
          acck += kout;                            // padded features=0, y=0
        }
      }
    }
    tsum += acck - (a0 * y0c + a1 * y1c);
  }

  // deterministic block tree-reduction
  red[tid] = tsum;
  __syncthreads();
  for (int s = 128; s > 0; s >>= 1) {
    if (tid < s) red[tid] += red[tid + s];
    __syncthreads();
  }
  if (tid == 0) {
    const int bid = (int)blockIdx.x + 16 * ((int)blockIdx.y + 8 * (int)blockIdx.z);
    ws[bid] = red[0];
  }
}

// deterministic final reduction of 512 block partials
__global__ __launch_bounds__(256) void gatedcrf_reduce(const float* __restrict__ ws,
                                                       float* __restrict__ out) {
  __shared__ float r[256];
  const int t = threadIdx.x;
  r[t] = ws[t] + ws[t + 256];
  __syncthreads();
  for (int s = 128; s > 0; s >>= 1) {
    if (t < s) r[t] += r[t + s];
    __syncthreads();
  }
  if (t == 0) out[0] = r[0] * (1.0f / (float)(N_ * D_ * H_ * W_));  // /524288
}

extern "C" void kernel_launch(void* const* d_in, const int* in_sizes, int n_in,
                              void* d_out, int out_size, void* d_ws, size_t ws_size,
                              hipStream_t stream) {
  (void)in_sizes; (void)n_in; (void)out_size; (void)ws_size;
  const float* y = (const float*)d_in[0];   // (2,2,64,64,64) f32
  const float* s = (const float*)d_in[1];   // (2,1,64,64,64) f32
  float* out = (float*)d_out;               // scalar f32
  float* ws  = (float*)d_ws;                // 512 block partials

  dim3 grid(W_ / TW, H_ / TH, N_ * 2);      // (16, 8, 4) = 512 blocks
  gatedcrf_kernel<<<grid, 256, 0, stream>>>(y, s, ws);
  gatedcrf_reduce<<<1, 256, 0, stream>>>(ws, out);
}